// SelfModifyingDeltaMemory_2534030704726
// MI455X (gfx1250) — compile-verified
//
#include <hip/hip_runtime.h>
#include <hip/hip_bf16.h>
#include <math.h>
#include <stdint.h>

// ---------------------------------------------------------------------------
// SelfModifyingDeltaMemory on MI455X (gfx1250)
//   B=4, S=512, N=4, D=256  ->  16 flat sequences x 512 tokens x 256 features
// Plan:
//   1) All token-parallel projections via fp32 WMMA (v_wmma_f32_16x16x4_f32).
//   2) Per-token gates (eta/alpha) + k-row L2 normalize in one small kernel.
//   3) Sequential delta-memory scan: one workgroup per head, M (256x256 f32)
//      resident in LDS (~271KB < 320KB/WGP), 512 steps, with per-step q/k/v
//      staged by double-buffered GLOBAL_LOAD_ASYNC_TO_LDS_B32 (ASYNCcnt).
//   4) Output projection (WMMA) + residual + layernorm.
// ---------------------------------------------------------------------------

typedef float v2f __attribute__((ext_vector_type(2)));
typedef float v8f __attribute__((ext_vector_type(8)));

#define TOK   8192          // 16 * 512 tokens
#define FD    256
#define SEQ   512
#define NHEAD 16

__device__ __forceinline__ float sigmoidf_(float x) {
  return 1.0f / (1.0f + __expf(-x));
}

// async global -> LDS copy, 4 bytes per enabled lane (CDNA5, ASYNCcnt-tracked)
__device__ __forceinline__ void async_ld_b32(float* ldsp, const float* g) {
  uint32_t off = (uint32_t)(uintptr_t)(__attribute__((address_space(3))) float*)ldsp;
  asm volatile("global_load_async_to_lds_b32 %0, %1, off"
               :: "v"(off), "v"(g) : "memory");
}
#define S_WAIT_ASYNC_LE3() asm volatile("s_wait_asynccnt 0x3" ::: "memory")
#define S_WAIT_ASYNC_0()   asm volatile("s_wait_asynccnt 0x0" ::: "memory")

// ---------------------------------------------------------------------------
// WMMA fp32 GEMM:  C[M,N] = act(A[M,K] @ W[K,N] + bias)
// One wave computes a 16(M) x 64(N) tile: 4 accumulators, A-frag reused 4x
// per K-step. K stepped by 4 (16x16x4 f32 WMMA).
// ---------------------------------------------------------------------------
template <int ACT>   // 0 = none, 1 = silu
__global__ __launch_bounds__(32)
void gemm_wmma_f32(const float* __restrict__ A, const float* __restrict__ W,
                   const float* __restrict__ bias, float* __restrict__ C,
                   int M, int N, int K) {
  const int nb   = N >> 6;                 // 64-wide N tiles
  const int bm   = blockIdx.x / nb;
  const int bn   = blockIdx.x % nb;
  const int lane = threadIdx.x & 31;
  const int half = lane >> 4;              // 0: lanes 0-15, 1: lanes 16-31
  const int l    = lane & 15;

  v8f acc[4] = {v8f{}, v8f{}, v8f{}, v8f{}};

  const float* Ar = A + (size_t)(bm * 16 + l) * K;
  const float* Wb = W + (size_t)(bn * 64 + l);

  for (int k0 = 0; k0 < K; k0 += 4) {
    v2f a;
    a.x = Ar[k0 + 2 * half];
    a.y = Ar[k0 + 2 * half + 1];

    const float* w0 = Wb + (size_t)(k0 + 2 * half) * N;
    const float* w1 = w0 + N;
    v2f b0 = {w0[0],  w1[0]};
    v2f b1 = {w0[16], w1[16]};
    v2f b2 = {w0[32], w1[32]};
    v2f b3 = {w0[48], w1[48]};

    acc[0] = __builtin_amdgcn_wmma_f32_16x16x4_f32(false, a, false, b0, (short)0, acc[0], false, false);
    acc[1] = __builtin_amdgcn_wmma_f32_16x16x4_f32(false, a, false, b1, (short)0, acc[1], false, false);
    acc[2] = __builtin_amdgcn_wmma_f32_16x16x4_f32(false, a, false, b2, (short)0, acc[2], false, false);
    acc[3] = __builtin_amdgcn_wmma_f32_16x16x4_f32(false, a, false, b3, (short)0, acc[3], false, false);
  }

  for (int j = 0; j < 4; ++j) {
    const int col = bn * 64 + j * 16 + l;
    const float bv = bias ? bias[col] : 0.0f;
    for (int i = 0; i < 8; ++i) {
      const int row = bm * 16 + 8 * half + i;
      float val = acc[j][i] + bv;
      if (ACT == 1) val = val * sigmoidf_(val);   // silu
      C[(size_t)row * N + col] = val;
    }
  }
}

// ---------------------------------------------------------------------------
// Per-token gates + k normalization.
// ---------------------------------------------------------------------------
__global__ __launch_bounds__(256)
void gate_norm_kernel(const float* __restrict__ he, const float* __restrict__ ha,
                      const float* __restrict__ We2, const float* __restrict__ be2,
                      const float* __restrict__ Wa2, const float* __restrict__ ba2,
                      float* __restrict__ k, float* __restrict__ eta,
                      float* __restrict__ alpha) {
  const int tok = blockIdx.x * blockDim.x + threadIdx.x;
  if (tok >= TOK) return;

  float se = 0.f, sa = 0.f;
  const float* hep = he + (size_t)tok * 64;
  const float* hap = ha + (size_t)tok * 64;
  for (int j = 0; j < 64; ++j) {
    se += hep[j] * We2[j];
    sa += hap[j] * Wa2[j];
  }
  eta[tok]   = 0.1f * sigmoidf_(se + be2[0]) + 0.01f;
  alpha[tok] = 0.5f * sigmoidf_(sa + ba2[0]) + 0.5f;

  float* kp = k + (size_t)tok * FD;
  float ss = 0.f;
  for (int j = 0; j < FD; ++j) { float kv = kp[j]; ss += kv * kv; }
  const float inv = 1.0f / fmaxf(sqrtf(ss), 1e-12f);
  for (int j = 0; j < FD; ++j) kp[j] *= inv;
}

// ---------------------------------------------------------------------------
// Sequential delta-memory scan. One workgroup (256 threads = 8 wave32) per
// head; M[256][256] f32 lives in LDS for all 512 steps (stride 258 padding
// -> conflict-free row access). q/k/v rows of step t+1 are staged into the
// alternate LDS buffer via async global->LDS loads while step t computes.
// Per step:
//   pass1 (fused): out_t[i] = sum_j M[i][j]*q[j],  Mk[i] = sum_j M[i][j]*kn[j]
//   pass2:         M[i][j]  = a*M[i][j] + e*(v[i]-Mk[i])*kn[j]
// ---------------------------------------------------------------------------
#define MSTRIDE 258
#define SCAN_LDS_FLOATS (FD * MSTRIDE + 7 * FD)   // M + 2x(q,k,v) + Mk

__global__ __launch_bounds__(256)
void scan_kernel(const float* __restrict__ q, const float* __restrict__ kn,
                 const float* __restrict__ vhat, const float* __restrict__ eta,
                 const float* __restrict__ alpha, float* __restrict__ out,
                 float* __restrict__ Mout) {
  extern __shared__ float lds[];
  float* Msh  = lds;                         // FD * MSTRIDE
  float* vecs = lds + FD * MSTRIDE;          // 2 x {q,k,v} x FD
  float* Mk   = vecs + 6 * FD;               // FD

  const int h   = blockIdx.x;                // head 0..15
  const int tid = threadIdx.x;               // 0..255 (row owned by thread)

  for (int j = tid; j < FD * MSTRIDE; j += 256) Msh[j] = 0.0f;

  const float* qh = q    + (size_t)h * SEQ * FD;
  const float* kh = kn   + (size_t)h * SEQ * FD;
  const float* vh = vhat + (size_t)h * SEQ * FD;
  float*       oh = out  + (size_t)h * SEQ * FD;

  // stage step 0 into buffer 0 (3 async instructions per wave)
  async_ld_b32(vecs + 0 * FD + tid, qh + tid);
  async_ld_b32(vecs + 1 * FD + tid, kh + tid);
  async_ld_b32(vecs + 2 * FD + tid, vh + tid);

  for (int t = 0; t < SEQ; ++t) {
    const int cur = (t & 1) * 3 * FD;
    const int nxt = ((t + 1) & 1) * 3 * FD;

    if (t + 1 < SEQ) {
      // prefetch step t+1 into the alternate buffer, then wait only for the
      // 3 older (step-t) async loads: async loads complete in order.
      const size_t o = (size_t)(t + 1) * FD + tid;
      async_ld_b32(vecs + nxt + 0 * FD + tid, qh + o);
      async_ld_b32(vecs + nxt + 1 * FD + tid, kh + o);
      async_ld_b32(vecs + nxt + 2 * FD + tid, vh + o);
      S_WAIT_ASYNC_LE3();
    } else {
      S_WAIT_ASYNC_0();
    }
    __syncthreads();                         // step-t data visible to all waves

    const float* qs = vecs + cur + 0 * FD;
    const float* ks = vecs + cur + 1 * FD;
    const float* vs = vecs + cur + 2 * FD;
    const float e = eta[h * SEQ + t];
    const float a = alpha[h * SEQ + t];

    // pass 1: fused dual matvec over row `tid` of M (one LDS read of M
    // services both reductions)
    const float2* Mrow = (const float2*)(Msh + (size_t)tid * MSTRIDE);
    const float2* q2 = (const float2*)qs;
    const float2* k2 = (const float2*)ks;
    float oq = 0.f, ok = 0.f;
    #pragma unroll 8
    for (int j = 0; j < FD / 2; ++j) {
      float2 m = Mrow[j];
      float2 qv = q2[j];
      float2 kv = k2[j];
      oq = fmaf(m.x, qv.x, fmaf(m.y, qv.y, oq));
      ok = fmaf(m.x, kv.x, fmaf(m.y, kv.y, ok));
    }
    oh[(size_t)t * FD + tid] = oq;           // out_t uses M_{t-1}
    Mk[tid] = ok;
    __syncthreads();

    // pass 2: rank-1 delta update of row `tid`
    const float coef = e * (vs[tid] - Mk[tid]);
    float2* Mrw = (float2*)(Msh + (size_t)tid * MSTRIDE);
    #pragma unroll 8
    for (int j = 0; j < FD / 2; ++j) {
      float2 m = Mrw[j];
      float2 kv = k2[j];
      m.x = fmaf(a, m.x, coef * kv.x);
      m.y = fmaf(a, m.y, coef * kv.y);
      Mrw[j] = m;
    }
    __syncthreads();                         // also guards nxt-buffer reuse
  }

  // emit final M (second output of the reference tuple)
  for (int idx = tid; idx < FD * FD; idx += 256) {
    const int r = idx >> 8, c = idx & 255;
    Mout[(size_t)h * FD * FD + idx] = Msh[(size_t)r * MSTRIDE + c];
  }
}

// ---------------------------------------------------------------------------
// Residual add + layernorm. One 256-thread block per token.
// ---------------------------------------------------------------------------
__global__ __launch_bounds__(256)
void ln_kernel(const float* __restrict__ ypre, const float* __restrict__ x,
               const float* __restrict__ gamma, const float* __restrict__ beta,
               float* __restrict__ yout) {
  __shared__ float red[256];
  const int tok = blockIdx.x;
  const int tid = threadIdx.x;

  const float val = ypre[(size_t)tok * FD + tid] + x[(size_t)tok * FD + tid];

  red[tid] = val;
  __syncthreads();
  for (int s = 128; s > 0; s >>= 1) {
    if (tid < s) red[tid] += red[tid + s];
    __syncthreads();
  }
  const float mu = red[0] * (1.0f / FD);
  __syncthreads();

  const float d = val - mu;
  red[tid] = d * d;
  __syncthreads();
  for (int s = 128; s > 0; s >>= 1) {
    if (tid < s) red[tid] += red[tid + s];
    __syncthreads();
  }
  const float var = red[0] * (1.0f / FD);

  yout[(size_t)tok * FD + tid] = d * rsqrtf(var + 1e-5f) * gamma[tid] + beta[tid];
}

// ---------------------------------------------------------------------------
// Launch. Input dict order:
//  0:x 1:Wq 2:Wk 3:Wv 4:Wg1 5:bg1 6:Wg2 7:bg2 8:We1 9:be1 10:We2 11:be2
// 12:Wa1 13:ba1 14:Wa2 15:ba2 16:Wo 17:bo 18:gamma 19:beta
// d_out: y (2,097,152 f32) then M (1,048,576 f32)
// ---------------------------------------------------------------------------
extern "C" void kernel_launch(void* const* d_in, const int* in_sizes, int n_in,
                              void* d_out, int out_size, void* d_ws, size_t ws_size,
                              hipStream_t stream) {
  const float* x   = (const float*)d_in[0];
  const float* Wq  = (const float*)d_in[1];
  const float* Wk  = (const float*)d_in[2];
  const float* Wv  = (const float*)d_in[3];
  const float* Wg1 = (const float*)d_in[4];
  const float* bg1 = (const float*)d_in[5];
  const float* Wg2 = (const float*)d_in[6];
  const float* bg2 = (const float*)d_in[7];
  const float* We1 = (const float*)d_in[8];
  const float* be1 = (const float*)d_in[9];
  const float* We2 = (const float*)d_in[10];
  const float* be2 = (const float*)d_in[11];
  const float* Wa1 = (const float*)d_in[12];
  const float* ba1 = (const float*)d_in[13];
  const float* Wa2 = (const float*)d_in[14];
  const float* ba2 = (const float*)d_in[15];
  const float* Wo  = (const float*)d_in[16];
  const float* bo  = (const float*)d_in[17];
  const float* gam = (const float*)d_in[18];
  const float* bet = (const float*)d_in[19];

  float* ws = (float*)d_ws;
  const size_t TF = (size_t)TOK * FD;       // 2,097,152
  float* q    = ws;                          // TF
  float* k    = ws + TF;                     // TF (kn after gate_norm)
  float* v    = ws + 2 * TF;                 // TF
  float* hg   = ws + 3 * TF;                 // TF
  float* vhat = ws + 4 * TF;                 // TF
  float* he   = ws + 5 * TF;                 // TOK*64
  float* ha   = he + (size_t)TOK * 64;       // TOK*64
  float* eta  = ha + (size_t)TOK * 64;       // TOK
  float* alp  = eta + TOK;                   // TOK
  float* out  = hg;                          // reuse: hg dead after vhat
  float* ypre = v;                           // reuse: v dead after hg

  float* y    = (float*)d_out;
  float* Mout = y + TF;

  const dim3 g256(TOK / 16 * (FD / 64));     // 2048 blocks, 256-wide output
  const dim3 g64 (TOK / 16 * (64 / 64));     // 512 blocks,  64-wide output
  const dim3 b32(32);

  // token-parallel projections (all fp32 WMMA)
  gemm_wmma_f32<0><<<g256, b32, 0, stream>>>(x,  Wq,  nullptr, q,    TOK, FD, FD);
  gemm_wmma_f32<0><<<g256, b32, 0, stream>>>(x,  Wk,  nullptr, k,    TOK, FD, FD);
  gemm_wmma_f32<0><<<g256, b32, 0, stream>>>(x,  Wv,  nullptr, v,    TOK, FD, FD);
  gemm_wmma_f32<1><<<g64,  b32, 0, stream>>>(x,  We1, be1,     he,   TOK, 64, FD);
  gemm_wmma_f32<1><<<g64,  b32, 0, stream>>>(x,  Wa1, ba1,     ha,   TOK, 64, FD);
  gemm_wmma_f32<1><<<g256, b32, 0, stream>>>(v,  Wg1, bg1,     hg,   TOK, FD, FD);
  gemm_wmma_f32<0><<<g256, b32, 0, stream>>>(hg, Wg2, bg2,     vhat, TOK, FD, FD);

  // gates + k normalization
  gate_norm_kernel<<<dim3(TOK / 256), dim3(256), 0, stream>>>(
      he, ha, We2, be2, Wa2, ba2, k, eta, alp);

  // sequential scan: 16 heads -> 16 WGPs, M resident in ~271KB LDS each,
  // async double-buffered q/k/v staging
  scan_kernel<<<dim3(NHEAD), dim3(256), SCAN_LDS_FLOATS * sizeof(float), stream>>>(
      q, k, vhat, eta, alp, out, Mout);

  // output projection + residual + layernorm
  gemm_wmma_f32<0><<<g256, b32, 0, stream>>>(out, Wo, bo, ypre, TOK, FD, FD);
  ln_kernel<<<dim3(TOK), dim3(256), 0, stream>>>(ypre, x, gam, bet, y);
}